// MambaBlock_54494545052144
// MI455X (gfx1250) — compile-verified
//
#include <hip/hip_runtime.h>
#include <hip/hip_bf16.h>
#include <math.h>

typedef __attribute__((ext_vector_type(2))) float v2f;
typedef __attribute__((ext_vector_type(8))) float v8f;

#define BATCH   4
#define DIM     128
#define DINNER  256
#define DSTATE  16
#define DTRANK  8
#define HH      56
#define WW      56
#define LSEQ    (HH*WW)        // 3136
#define MROWS   (BATCH*LSEQ)   // 12544

__device__ __forceinline__ float siluf(float x) { return x / (1.f + expf(-x)); }

// ---------------------------------------------------------------------------
// conv1: 3x3, 64 -> 128, pad 1, writes token-major (B, L, 128)
// ---------------------------------------------------------------------------
__global__ void conv1_kernel(const float* __restrict__ x,
                             const float* __restrict__ cw,
                             const float* __restrict__ cb,
                             float* __restrict__ xf) {
    int p  = blockIdx.x;             // b*L + l
    int b  = p / LSEQ;
    int l  = p - b * LSEQ;
    int h  = l / WW, w = l - h * WW;
    int oc = threadIdx.x;            // 0..127
    float acc = cb[oc];
    for (int ic = 0; ic < 64; ++ic) {
        const float* xp = x + ((size_t)(b * 64 + ic) * HH) * WW;
        const float* wp = cw + ((size_t)oc * 64 + ic) * 9;
        #pragma unroll
        for (int kh = 0; kh < 3; ++kh) {
            int hh = h + kh - 1;
            if ((unsigned)hh >= HH) continue;
            #pragma unroll
            for (int kw = 0; kw < 3; ++kw) {
                int ww2 = w + kw - 1;
                if ((unsigned)ww2 >= WW) continue;
                acc += xp[hh * WW + ww2] * wp[kh * 3 + kw];
            }
        }
    }
    xf[(size_t)p * DIM + oc] = acc;
}

// ---------------------------------------------------------------------------
// LayerNorm over last dim (128). One wave per row, 4 elems per lane.
// ---------------------------------------------------------------------------
__global__ void ln_kernel(const float* __restrict__ X,
                          const float* __restrict__ w,
                          const float* __restrict__ b,
                          float* __restrict__ Y, int rows) {
    int wave = threadIdx.x >> 5, lane = threadIdx.x & 31;
    int row = blockIdx.x * 8 + wave;
    if (row >= rows) return;
    const float4* xp = (const float4*)(X + (size_t)row * DIM);
    float4 v = xp[lane];
    float s = v.x + v.y + v.z + v.w;
    #pragma unroll
    for (int off = 16; off; off >>= 1) s += __shfl_xor(s, off, 32);
    float mu = s * (1.f / DIM);
    float d0 = v.x - mu, d1 = v.y - mu, d2 = v.z - mu, d3 = v.w - mu;
    float q = d0 * d0 + d1 * d1 + d2 * d2 + d3 * d3;
    #pragma unroll
    for (int off = 16; off; off >>= 1) q += __shfl_xor(q, off, 32);
    float inv = rsqrtf(q * (1.f / DIM) + 1e-5f);
    int c = lane * 4;
    float4 o;
    o.x = d0 * inv * w[c + 0] + b[c + 0];
    o.y = d1 * inv * w[c + 1] + b[c + 1];
    o.z = d2 * inv * w[c + 2] + b[c + 2];
    o.w = d3 * inv * w[c + 3] + b[c + 3];
    ((float4*)(Y + (size_t)row * DIM))[lane] = o;
}

// ---------------------------------------------------------------------------
// WMMA f32 GEMM: C[M,N] = A[M,K] @ W[N,K]^T  (+ epilogues)
// One wave per 16x64 output strip (4 N-tiles), V_WMMA_F32_16X16X4_F32 along K.
// A fragment loaded once per k-step, reused across the 4 tiles.
// A-lane layout: lane = half*16 + m, a = {A[m][k+2*half], A[m][k+2*half+1]}
// B-lane layout mirrors with n in place of m -> W rows load identically.
// D layout: lane(half,mr): acc[i] = C[half*8+i][mr]
// ---------------------------------------------------------------------------
#define E_NONE      0
#define E_ADD       1
#define E_BIAS_GELU 2
#define E_BIAS_ADD2 3
#define NT          4           // N-tiles per wave (16x64 strip)

template <int E>
__global__ void gemm_wmma(const float* __restrict__ A,
                          const float* __restrict__ W,
                          const float* __restrict__ bias,
                          const float* __restrict__ add,
                          float* __restrict__ C,
                          float* __restrict__ C2,
                          int M, int N, int K) {
    int wid     = blockIdx.x * (blockDim.x >> 5) + (threadIdx.x >> 5);
    int stripsN = N >> 6;                        // strips of 64 columns
    int tm = wid / stripsN, tn = wid - tm * stripsN;
    if (tm * 16 >= M) return;
    int lane = threadIdx.x & 31;
    int half = lane >> 4, mr = lane & 15;
    const float* Ap  = A + (size_t)(tm * 16 + mr) * K + half * 2;
    const float* Wp0 = W + (size_t)(tn * 64 + mr) * K + half * 2;
    v8f acc0 = {}, acc1 = {}, acc2 = {}, acc3 = {};
    #pragma unroll 2
    for (int k = 0; k < K; k += 4) {
        v2f a  = *(const v2f*)(Ap + k);
        v2f b0 = *(const v2f*)(Wp0 + k);
        v2f b1 = *(const v2f*)(Wp0 + (size_t)16 * K + k);
        v2f b2 = *(const v2f*)(Wp0 + (size_t)32 * K + k);
        v2f b3 = *(const v2f*)(Wp0 + (size_t)48 * K + k);
        acc0 = __builtin_amdgcn_wmma_f32_16x16x4_f32(false, a, false, b0, (short)0, acc0, false, false);
        acc1 = __builtin_amdgcn_wmma_f32_16x16x4_f32(false, a, false, b1, (short)0, acc1, false, false);
        acc2 = __builtin_amdgcn_wmma_f32_16x16x4_f32(false, a, false, b2, (short)0, acc2, false, false);
        acc3 = __builtin_amdgcn_wmma_f32_16x16x4_f32(false, a, false, b3, (short)0, acc3, false, false);
    }
    int row0 = tm * 16 + half * 8;
    v8f accs[NT] = {acc0, acc1, acc2, acc3};
    #pragma unroll
    for (int t = 0; t < NT; ++t) {
        int col = tn * 64 + t * 16 + mr;
        #pragma unroll
        for (int i = 0; i < 8; ++i) {
            float v = accs[t][i];
            size_t idx = (size_t)(row0 + i) * N + col;
            if (E == E_ADD)       v += add[idx];
            if (E == E_BIAS_GELU) { v += bias[col]; v = 0.5f * v * (1.f + erff(v * 0.70710678118654752f)); }
            if (E == E_BIAS_ADD2) { v += bias[col]; C2[idx] = add[idx] + v; }
            C[idx] = v;
        }
    }
}

// ---------------------------------------------------------------------------
// Depthwise causal conv1d (kernel 4, pad (3,0)) over L + SiLU.
// xi = XZ[..., :256]; writes U (B,L,256)
// ---------------------------------------------------------------------------
__global__ void dwconv_silu_kernel(const float* __restrict__ XZ,
                                   const float* __restrict__ w1d,
                                   const float* __restrict__ b1d,
                                   float* __restrict__ U) {
    int idx = blockIdx.x * blockDim.x + threadIdx.x;   // B*L*256
    int d  = idx & (DINNER - 1);
    int bl = idx >> 8;
    int b  = bl / LSEQ;
    int l  = bl - b * LSEQ;
    float acc = b1d[d];
    #pragma unroll
    for (int j = 0; j < 4; ++j) {
        int ls = l - 3 + j;
        if (ls >= 0)
            acc += XZ[((size_t)(b * LSEQ + ls)) * (2 * DINNER) + d] * w1d[d * 4 + j];
    }
    U[(size_t)bl * DINNER + d] = siluf(acc);
}

// ---------------------------------------------------------------------------
// x_proj: XDBL[row, j] = dot(U[row,:256], x_proj_w[j,:256]), j < 40.
// One block per row; U row staged through LDS.
// ---------------------------------------------------------------------------
__global__ void xproj_kernel(const float* __restrict__ U,
                             const float* __restrict__ xw,
                             float* __restrict__ XDBL) {
    __shared__ float urow[DINNER];
    int row = blockIdx.x;
    const float* up = U + (size_t)row * DINNER;
    for (int k = threadIdx.x; k < DINNER; k += blockDim.x) urow[k] = up[k];
    __syncthreads();
    int j = threadIdx.x;
    if (j < DTRANK + 2 * DSTATE) {
        const float* wp = xw + (size_t)j * DINNER;
        float acc = 0.f;
        #pragma unroll 8
        for (int k = 0; k < DINNER; ++k) acc += urow[k] * wp[k];
        XDBL[(size_t)row * 40 + j] = acc;
    }
}

// ---------------------------------------------------------------------------
// dt = softplus(XDBL[:, :8] @ dt_proj_w^T + dt_proj_b)  (B,L,256)
// ---------------------------------------------------------------------------
__global__ void dt_kernel(const float* __restrict__ XDBL,
                          const float* __restrict__ dw,
                          const float* __restrict__ db,
                          float* __restrict__ DT) {
    int idx = blockIdx.x * blockDim.x + threadIdx.x;   // B*L*256
    int d   = idx & (DINNER - 1);
    int row = idx >> 8;
    const float* xp = XDBL + (size_t)row * 40;
    const float* wp = dw + (size_t)d * DTRANK;
    float acc = db[d];
    #pragma unroll
    for (int j = 0; j < DTRANK; ++j) acc += xp[j] * wp[j];
    float sp = (acc > 20.f) ? acc : log1pf(expf(acc));
    DT[(size_t)row * DINNER + d] = sp;
}

// ---------------------------------------------------------------------------
// Selective scan. 16 lanes per (b,d) channel, one state per lane.
// y = (sum_s h_s * C_s + u*Dp) * silu(z);  h_s = exp(dt*A_s)*h_s + dt*B_s*u
// ---------------------------------------------------------------------------
__global__ void scan_kernel(const float* __restrict__ DT,
                            const float* __restrict__ U,
                            const float* __restrict__ XDBL,
                            const float* __restrict__ XZ,
                            const float* __restrict__ A_log,
                            const float* __restrict__ Dp,
                            float* __restrict__ YS) {
    int gid = blockIdx.x * blockDim.x + threadIdx.x;   // 16384 threads
    int s  = gid & (DSTATE - 1);
    int ch = gid >> 4;                                 // 0..1023
    int d  = ch & (DINNER - 1);
    int b  = ch >> 8;
    float Aval = -expf(A_log[d * DSTATE + s]);
    float dpv  = Dp[d];
    float h = 0.f;
    for (int l = 0; l < LSEQ; ++l) {
        size_t rl = (size_t)(b * LSEQ + l);
        float dt = DT[rl * DINNER + d];
        float uu = U[rl * DINNER + d];
        float Bs = XDBL[rl * 40 + DTRANK + s];
        float Cs = XDBL[rl * 40 + DTRANK + DSTATE + s];
        h = expf(dt * Aval) * h + dt * Bs * uu;
        float y = h * Cs;
        #pragma unroll
        for (int off = 8; off; off >>= 1) y += __shfl_xor(y, off, 16);
        if (s == 0) {
            float z = XZ[rl * (2 * DINNER) + DINNER + d];
            YS[rl * DINNER + d] = (y + uu * dpv) * siluf(z);
        }
    }
}

// ---------------------------------------------------------------------------
// Transpose (B, L, C) -> (B, C, L) via LDS 32x32 tiles (L = 3136 = 98*32)
// ---------------------------------------------------------------------------
__global__ void transpose_kernel(const float* __restrict__ RN,
                                 float* __restrict__ out) {
    __shared__ float tile[32][33];
    int b  = blockIdx.z;
    int l0 = blockIdx.x * 32, c0 = blockIdx.y * 32;
    int tx = threadIdx.x, ty = threadIdx.y;   // 32 x 8
    for (int r = ty; r < 32; r += 8)
        tile[r][tx] = RN[((size_t)b * LSEQ + (l0 + r)) * DIM + (c0 + tx)];
    __syncthreads();
    for (int r = ty; r < 32; r += 8)
        out[((size_t)b * DIM + (c0 + r)) * LSEQ + (l0 + tx)] = tile[tx][r];
}

// ---------------------------------------------------------------------------
// Launch
// ---------------------------------------------------------------------------
extern "C" void kernel_launch(void* const* d_in, const int* in_sizes, int n_in,
                              void* d_out, int out_size, void* d_ws, size_t ws_size,
                              hipStream_t stream) {
    const float* x        = (const float*)d_in[0];
    const float* conv1_w  = (const float*)d_in[1];
    const float* conv1_b  = (const float*)d_in[2];
    const float* norm1_w  = (const float*)d_in[3];
    const float* norm1_b  = (const float*)d_in[4];
    const float* norm2_w  = (const float*)d_in[5];
    const float* norm2_b  = (const float*)d_in[6];
    const float* in_proj_w = (const float*)d_in[7];
    const float* conv1d_w  = (const float*)d_in[8];
    const float* conv1d_b  = (const float*)d_in[9];
    const float* x_proj_w  = (const float*)d_in[10];
    const float* dt_proj_w = (const float*)d_in[11];
    const float* dt_proj_b = (const float*)d_in[12];
    const float* A_log     = (const float*)d_in[13];
    const float* Dp        = (const float*)d_in[14];
    const float* out_proj_w = (const float*)d_in[15];
    const float* fc1_w = (const float*)d_in[16];
    const float* fc1_b = (const float*)d_in[17];
    const float* fc2_w = (const float*)d_in[18];
    const float* fc2_b = (const float*)d_in[19];

    float* ws = (float*)d_ws;
    const size_t SZ_TOK128 = (size_t)MROWS * DIM;        // 1,605,632
    const size_t SZ_TOK256 = (size_t)MROWS * DINNER;     // 3,211,264
    const size_t SZ_TOK512 = (size_t)MROWS * 2 * DINNER; // 6,422,528
    const size_t SZ_TOK40  = (size_t)MROWS * 40;         //   501,760
    float* XF   = ws;                 ws += SZ_TOK128;   // x_flat
    float* XN   = ws;                 ws += SZ_TOK128;   // LN output (both norms)
    float* XZ   = ws;                 ws += SZ_TOK512;   // in_proj out (xi|z)
    float* U    = ws;                 ws += SZ_TOK256;   // silu(conv1d)
    float* XDBL = ws;                 ws += SZ_TOK40;    // x_proj out
    float* DT   = ws;                 ws += SZ_TOK256;   // softplus dt
    float* YS   = ws;                 ws += SZ_TOK256;   // scan out; reused as gelu(fc1)
    float* R    = ws;                 ws += SZ_TOK128;   // residual_output
    float* RN   = ws;                 ws += SZ_TOK128;   // residual_norm

    dim3 blk256(256);
    // conv1 -> XF token-major
    conv1_kernel<<<dim3(MROWS), dim3(DIM), 0, stream>>>(x, conv1_w, conv1_b, XF);

    for (int it = 0; it < 3; ++it) {
        // LN1
        ln_kernel<<<dim3(MROWS / 8), blk256, 0, stream>>>(XF, norm1_w, norm1_b, XN, MROWS);
        // in_proj: (12544 x 512) = XN(12544x128) @ W(512x128)^T
        {
            int waves = (MROWS / 16) * ((2 * DINNER) / 64);   // 784 * 8
            gemm_wmma<E_NONE><<<dim3((waves + 7) / 8), blk256, 0, stream>>>(
                XN, in_proj_w, nullptr, nullptr, XZ, nullptr, MROWS, 2 * DINNER, DIM);
        }
        // depthwise conv1d + silu
        dwconv_silu_kernel<<<dim3((MROWS * DINNER) / 256), blk256, 0, stream>>>(
            XZ, conv1d_w, conv1d_b, U);
        // x_proj
        xproj_kernel<<<dim3(MROWS), dim3(64), 0, stream>>>(U, x_proj_w, XDBL);
        // dt
        dt_kernel<<<dim3((MROWS * DINNER) / 256), blk256, 0, stream>>>(
            XDBL, dt_proj_w, dt_proj_b, DT);
        // selective scan (+ *silu(z), +u*Dp)
        scan_kernel<<<dim3((BATCH * DINNER * DSTATE) / 256), blk256, 0, stream>>>(
            DT, U, XDBL, XZ, A_log, Dp, YS);
        // out_proj + residual: R = XF + YS @ out_proj_w^T
        {
            int waves = (MROWS / 16) * (DIM / 64);            // 784 * 2
            gemm_wmma<E_ADD><<<dim3((waves + 7) / 8), blk256, 0, stream>>>(
                YS, out_proj_w, nullptr, XF, R, nullptr, MROWS, DIM, DINNER);
        }
        // LN2
        ln_kernel<<<dim3(MROWS / 8), blk256, 0, stream>>>(R, norm2_w, norm2_b, XN, MROWS);
        // fc1 + bias + exact gelu -> YS (reuse)
        {
            int waves = (MROWS / 16) * (DIM / 64);
            gemm_wmma<E_BIAS_GELU><<<dim3((waves + 7) / 8), blk256, 0, stream>>>(
                XN, fc1_w, fc1_b, nullptr, YS, nullptr, MROWS, DIM, DIM);
        }
        // fc2 + bias -> RN ; XF_next = R + RN
        {
            int waves = (MROWS / 16) * (DIM / 64);
            gemm_wmma<E_BIAS_ADD2><<<dim3((waves + 7) / 8), blk256, 0, stream>>>(
                YS, fc2_w, fc2_b, R, RN, XF, MROWS, DIM, DIM);
        }
    }
    // out = RN transposed to (B, C, L)
    transpose_kernel<<<dim3(LSEQ / 32, DIM / 32, BATCH), dim3(32, 8), 0, stream>>>(
        RN, (float*)d_out);
}